// DeformableConv2d_79156247265841
// MI455X (gfx1250) — compile-verified
//
#include <hip/hip_runtime.h>
#include <hip/hip_bf16.h>

typedef float v2f __attribute__((ext_vector_type(2)));
typedef float v4f __attribute__((ext_vector_type(4)));
typedef float v8f __attribute__((ext_vector_type(8)));
typedef int   v4i __attribute__((ext_vector_type(4)));

#define CIN   256
#define COUT  256
#define Hh    80
#define Ww    80
#define HW    6400
#define NB    4
#define NPOS  (NB * HW)     // 25600
#define BM    64
#define BN    128
#define LDK   36            // padded K stride (floats) in LDS: keeps v4f stores 16B aligned
#define NCHUNK 72           // 9 taps * 8 c-chunks of 32

// ---------------------------------------------------------------------------
// Kernel 1: offsets = conv3x3(x, offset_w) + offset_b   -> [B, 18, H, W]
// ---------------------------------------------------------------------------
__global__ __launch_bounds__(256) void offset_conv_kernel(
    const float* __restrict__ x, const float* __restrict__ ow,
    const float* __restrict__ ob, float* __restrict__ offs)
{
    __shared__ float wsm[162];                 // 18 ch * 9 taps for one cin
    const int tid = threadIdx.x;
    const int gp  = blockIdx.x * 256 + tid;    // flattened (b, pix); 6400 % 256 == 0
    const int b   = (blockIdx.x * 256) / HW;   // uniform per block
    const int pix = gp - b * HW;
    const int y   = pix / Ww;
    const int xx  = pix % Ww;

    float acc[18];
    #pragma unroll
    for (int ch = 0; ch < 18; ++ch) acc[ch] = ob[ch];

    const float* xb = x + (size_t)b * CIN * HW;
    for (int c = 0; c < CIN; ++c) {
        __syncthreads();
        if (tid < 162) {
            const int ch = tid / 9, tap = tid % 9;
            wsm[tid] = ow[((size_t)(ch * CIN + c)) * 9 + tap];
        }
        __syncthreads();
        float xv[9];
        const float* xc = xb + (size_t)c * HW;
        #pragma unroll
        for (int tap = 0; tap < 9; ++tap) {
            const int yy  = y  + tap / 3 - 1;
            const int xx2 = xx + tap % 3 - 1;
            xv[tap] = (yy >= 0 && yy < Hh && xx2 >= 0 && xx2 < Ww)
                          ? xc[yy * Ww + xx2] : 0.0f;
        }
        #pragma unroll
        for (int ch = 0; ch < 18; ++ch) {
            float a = acc[ch];
            #pragma unroll
            for (int tap = 0; tap < 9; ++tap) a += wsm[ch * 9 + tap] * xv[tap];
            acc[ch] = a;
        }
    }
    for (int ch = 0; ch < 18; ++ch)
        offs[((size_t)(b * 18 + ch)) * HW + pix] = acc[ch];
}

// ---------------------------------------------------------------------------
// Kernel 2: per (b, tap, pixel): 4 clamped gather indices + 4 masked weights
// ---------------------------------------------------------------------------
__global__ __launch_bounds__(256) void interp_precompute_kernel(
    const float* __restrict__ offs, v4i* __restrict__ tb_idx,
    v4f* __restrict__ tb_w)
{
    const int i = blockIdx.x * 256 + threadIdx.x;
    if (i >= NB * 9 * HW) return;
    const int b   = i / (9 * HW);
    const int r   = i % (9 * HW);
    const int t   = r / HW;
    const int pix = r % HW;
    const int y   = pix / Ww;
    const int xx  = pix % Ww;

    const float dy = offs[((size_t)(b * 18 + 2 * t))     * HW + pix];
    const float dx = offs[((size_t)(b * 18 + 2 * t + 1)) * HW + pix];
    const float py = (float)(y  + t / 3 - 1) + dy;
    const float px = (float)(xx + t % 3 - 1) + dx;
    const float y0f = floorf(py), x0f = floorf(px);
    const float wy = py - y0f,    wx = px - x0f;
    const int y0 = (int)y0f, x0 = (int)x0f;
    const int y1 = y0 + 1,   x1 = x0 + 1;

    auto valid = [](int yy, int xc) {
        return (yy >= 0 && yy < Hh && xc >= 0 && xc < Ww) ? 1.0f : 0.0f;
    };
    auto cl = [](int v, int hi) { return v < 0 ? 0 : (v > hi ? hi : v); };

    v4f w; v4i id;
    w.x = (1.0f - wy) * (1.0f - wx) * valid(y0, x0);
    w.y = (1.0f - wy) * wx          * valid(y0, x1);
    w.z = wy          * (1.0f - wx) * valid(y1, x0);
    w.w = wy          * wx          * valid(y1, x1);
    id.x = cl(y0, Hh - 1) * Ww + cl(x0, Ww - 1);
    id.y = cl(y0, Hh - 1) * Ww + cl(x1, Ww - 1);
    id.z = cl(y1, Hh - 1) * Ww + cl(x0, Ww - 1);
    id.w = cl(y1, Hh - 1) * Ww + cl(x1, Ww - 1);
    tb_w[i]   = w;
    tb_idx[i] = id;
}

// ---------------------------------------------------------------------------
// Kernel 2b: pack weights K-pair-major for coalesced b64 WMMA A-fragment loads
//   A2[(t*128 + c/2) * 256 + o] = { conv_w[o, c, t], conv_w[o, c+1, t] }
// ---------------------------------------------------------------------------
__global__ __launch_bounds__(256) void weight_pack_kernel(
    const float* __restrict__ cw, v2f* __restrict__ A2)
{
    const int i = blockIdx.x * 256 + threadIdx.x;   // over 1152*256
    if (i >= 1152 * 256) return;
    const int q  = i >> 8;          // pair index: t*128 + c2
    const int o  = i & 255;
    const int t  = q / 128;
    const int c  = (q % 128) * 2;
    v2f v;
    v.x = cw[((size_t)o * CIN + c)     * 9 + t];
    v.y = cw[((size_t)o * CIN + c + 1) * 9 + t];
    A2[i] = v;
}

// ---------------------------------------------------------------------------
// Kernel 3: D[o,p] = sum_k' A'[o,k'] * val[k',p] + bias   via f32 WMMA
//   k' = t*256 + c (tap-major). Block tile 64(M) x 128(N); wave tile 32x32.
//   Double-buffered LDS slab Bs[p][k] (K-contiguous -> single ds_load_b64 frags,
//   v4f build stores), software-pipelined gather/WMMA.
// ---------------------------------------------------------------------------
__global__ __launch_bounds__(256) void deform_gemm_kernel(
    const float* __restrict__ x, const v2f* __restrict__ A2,
    const float* __restrict__ cb, const v4i* __restrict__ tb_idx,
    const v4f* __restrict__ tb_w, float* __restrict__ out)
{
    __shared__ float Bs[2][BN * LDK];         // 2 x 18.4 KB

    const int tid  = threadIdx.x;
    const int lane = tid & 31;
    const int wave = tid >> 5;
    const int wm   = wave >> 2;               // 0..1
    const int wn   = wave & 3;                // 0..3
    const int M0   = blockIdx.y * BM;
    const int P0   = blockIdx.x * BN;
    const int b    = P0 / HW;                 // uniform (6400 % 128 == 0)
    const int pix  = (P0 % HW) + (tid & 127);

    const int p_local = tid & 127;            // N column this thread builds
    const int crow0   = (tid >> 7) * 16;      // 0 or 16: K rows this thread builds
    const int mrow    = lane & 15;            // M (or N) index within fragment
    const int khalf   = lane >> 4;            // 0 or 1: K pairing half

    const v8f vzero = {0.f,0.f,0.f,0.f,0.f,0.f,0.f,0.f};
    v8f acc[2][2];
    acc[0][0] = vzero; acc[0][1] = vzero; acc[1][0] = vzero; acc[1][1] = vzero;

    const float* xb = x + (size_t)b * CIN * HW;

    auto build = [&](int chunk, int bufi) {
        const int t     = chunk >> 3;
        const int cbase = (chunk & 7) << 5;
        const size_t te = (size_t)(b * 9 + t) * HW + pix;
        const v4i id    = tb_idx[te];
        const v4f w4    = tb_w[te];
        float v[16];
        #pragma unroll
        for (int i = 0; i < 16; ++i) {
            const float* row = xb + (size_t)(cbase + crow0 + i) * HW;
            v[i] = w4.x * row[id.x] + w4.y * row[id.y]
                 + w4.z * row[id.z] + w4.w * row[id.w];
        }
        float* dst = &Bs[bufi][p_local * LDK + crow0];
        #pragma unroll
        for (int j = 0; j < 4; ++j) {
            v4f t4 = { v[4*j], v[4*j+1], v[4*j+2], v[4*j+3] };
            *(v4f*)(dst + 4 * j) = t4;
        }
    };

    build(0, 0);
    int bufi = 0;
    for (int chunk = 0; chunk < NCHUNK; ++chunk) {
        const int t     = chunk >> 3;
        const int cbase = (chunk & 7) << 5;
        __syncthreads();                      // slab[bufi] ready; slab[bufi^1] free

        // A fragments: coalesced global_load_b64 from packed weights
        v2f afrag[2][8];
        #pragma unroll
        for (int mt = 0; mt < 2; ++mt) {
            const int o = M0 + wm * 32 + mt * 16 + mrow;
            const v2f* arow = A2 + (size_t)(t * 128 + (cbase >> 1)) * COUT + o;
            #pragma unroll
            for (int s = 0; s < 8; ++s)
                afrag[mt][s] = arow[(size_t)(s * 2 + khalf) * COUT];
        }

        // overlap: gather next slab into the other buffer
        if (chunk + 1 < NCHUNK) build(chunk + 1, bufi ^ 1);

        // 8 K-steps of 4, 2x2 C tiles, B frags as single aligned ds_load_b64
        const float* Bp = Bs[bufi];
        #pragma unroll
        for (int s = 0; s < 8; ++s) {
            const int krow = s * 4 + khalf * 2;
            v2f bfrag[2];
            #pragma unroll
            for (int nt = 0; nt < 2; ++nt) {
                const int col = wn * 32 + nt * 16 + mrow;
                bfrag[nt] = *(const v2f*)&Bp[col * LDK + krow];
            }
            #pragma unroll
            for (int mt = 0; mt < 2; ++mt)
                #pragma unroll
                for (int nt = 0; nt < 2; ++nt)
                    acc[mt][nt] = __builtin_amdgcn_wmma_f32_16x16x4_f32(
                        false, afrag[mt][s], false, bfrag[nt],
                        (short)0, acc[mt][nt], false, false);
        }
        bufi ^= 1;
    }

    // ---- epilogue: C layout VGPR r -> M = r + 8*khalf ; N = lane%16
    #pragma unroll
    for (int mt = 0; mt < 2; ++mt) {
        const int obase = M0 + wm * 32 + mt * 16 + 8 * khalf;
        #pragma unroll
        for (int nt = 0; nt < 2; ++nt) {
            const int p  = P0 + wn * 32 + nt * 16 + mrow;
            const int px = p - b * HW;
            #pragma unroll
            for (int r = 0; r < 8; ++r) {
                const int o = obase + r;
                out[((size_t)(b * COUT + o)) * HW + px] = acc[mt][nt][r] + cb[o];
            }
        }
    }
}

// ---------------------------------------------------------------------------
extern "C" void kernel_launch(void* const* d_in, const int* in_sizes, int n_in,
                              void* d_out, int out_size, void* d_ws, size_t ws_size,
                              hipStream_t stream) {
    const float* x        = (const float*)d_in[0];   // [4,256,80,80]
    const float* offset_w = (const float*)d_in[1];   // [18,256,3,3]
    const float* offset_b = (const float*)d_in[2];   // [18]
    const float* conv_w   = (const float*)d_in[3];   // [256,256,3,3]
    const float* conv_b   = (const float*)d_in[4];   // [256]
    float*       out      = (float*)d_out;           // [4,256,80,80]

    // workspace layout (all 16B aligned):
    //   offs   : 4*18*6400 f32      = 1,843,200 B
    //   tb_idx : 4*9*6400 int4      = 3,686,400 B
    //   tb_w   : 4*9*6400 float4    = 3,686,400 B
    //   A2     : 1152*256 float2    = 2,359,296 B   (total ~11.6 MB)
    char* ws = (char*)d_ws;
    float* offs  = (float*)ws;
    v4i* tb_idx  = (v4i*)(ws + 1843200);
    v4f* tb_w    = (v4f*)(ws + 1843200 + 3686400);
    v2f* A2      = (v2f*)(ws + 1843200 + 3686400 + 3686400);

    offset_conv_kernel<<<NPOS / 256, 256, 0, stream>>>(x, offset_w, offset_b, offs);
    interp_precompute_kernel<<<(NB * 9 * HW + 255) / 256, 256, 0, stream>>>(offs, tb_idx, tb_w);
    weight_pack_kernel<<<(1152 * 256 + 255) / 256, 256, 0, stream>>>(conv_w, A2);
    deform_gemm_kernel<<<dim3(NPOS / BN, COUT / BM), 256, 0, stream>>>(
        x, A2, conv_b, tb_idx, tb_w, out);
}